// GIN_Classic_21449066676826
// MI455X (gfx1250) — compile-verified
//
#include <hip/hip_runtime.h>
#include <hip/hip_bf16.h>

#define N_NODES    50000
#define N_EDGES    800000
#define IN_CH      128
#define HID        256
#define OUT_CH     10
#define NUM_LAYERS 4
#define NUM_GRAPHS 128
#define BN_EPS     1e-5f

typedef float v2f __attribute__((ext_vector_type(2)));
typedef float v8f __attribute__((ext_vector_type(8)));

// ---------------------------------------------------------------- utilities
__global__ void k_zero(float* __restrict__ p, int n) {
    int i = blockIdx.x * blockDim.x + threadIdx.x;
    if (i < n) p[i] = 0.0f;
}

__global__ void k_copy(const float* __restrict__ src, float* __restrict__ dst, int n4) {
    int i = blockIdx.x * blockDim.x + threadIdx.x;
    if (i < n4) ((float4*)dst)[i] = ((const float4*)src)[i];
}

// --------------------------------------------------- GIN aggregation (edges)
// z[dst[e], :] += h[src[e], :]   (z pre-initialized to h => z = h + agg)
__global__ void k_edge_scatter(const float* __restrict__ h,
                               const int* __restrict__ src,
                               const int* __restrict__ dst,
                               float* __restrict__ z,
                               int C4 /* channels/4 */) {
    int tid = blockIdx.x * blockDim.x + threadIdx.x;
    int e  = tid / C4;
    int c4 = tid - e * C4;
    if (e >= N_EDGES) return;
    int s = src[e];
    int d = dst[e];
    int C = C4 * 4;
    float4 v = ((const float4*)(h + (long)s * C))[c4];
    float* zp = z + (long)d * C + c4 * 4;
    atomicAdd(zp + 0, v.x);
    atomicAdd(zp + 1, v.y);
    atomicAdd(zp + 2, v.z);
    atomicAdd(zp + 3, v.w);
}

// --------------------------------------------- W repack for WMMA B fragments
// Wq (float2) layout: idx = kpair*256 + wave*64 + m*4 + j
//   value = { W[2*kpair, n], W[2*kpair+1, n] },  n = wave*64 + j*16 + m
// A lane (ksel, m) of wave w then reads its 4 B-tile fragments for one K-step
// as two contiguous float4 (global_load_b128) at
//   (float4*)Wq + (ksel*128 + w*32 + m*2) + 256*s  (+0, +1)
__global__ void k_packW(const float* __restrict__ W, float2* __restrict__ Wq, int Khalf256) {
    int idx = blockIdx.x * blockDim.x + threadIdx.x;   // over (K/2)*256
    if (idx >= Khalf256) return;
    int kpair = idx >> 8;
    int r     = idx & 255;
    int wave  = r >> 6;
    int m     = (r >> 2) & 15;
    int j     = r & 3;
    int n     = wave * 64 + j * 16 + m;
    Wq[idx] = make_float2(W[(2 * kpair) * HID + n], W[(2 * kpair + 1) * HID + n]);
}

// ------------------------------------------------------ fp32 WMMA GEMM
// out[M x 256] = A[M x K] @ W[K x 256] + bias, lda == K (compile-time).
// block = 128 threads (4 wave32s); wave w: rows [16*bx,+16), cols [64*w,+64).
template <int K>
__global__ void k_gemm_wmma(const float* __restrict__ A,
                            const float2* __restrict__ Wq,
                            const float* __restrict__ bias,
                            float* __restrict__ out) {
    const int wave = threadIdx.x >> 5;
    const int lane = threadIdx.x & 31;
    const int m    = lane & 15;      // A row within tile / B-and-D column
    const int ksel = lane >> 4;      // 0: K pair {0,1}; 1: K pair {2,3}
    const int n0   = wave * 64;
    const int row  = blockIdx.x * 16 + m;

    v8f acc0 = {}, acc1 = {}, acc2 = {}, acc3 = {};
    const float*  Ap = A + (long)row * K + 2 * ksel;
    const float4* Bp = (const float4*)Wq + (ksel * 128 + wave * 32 + m * 2);

#pragma unroll 8
    for (int s = 0; s < K / 4; ++s) {
        float2 av = *(const float2*)(Ap + 4 * s);        // global_load_b64
        float4 q0 = Bp[256 * s];                         // global_load_b128
        float4 q1 = Bp[256 * s + 1];                     // global_load_b128
        v2f a;  a.x  = av.x; a.y  = av.y;
        v2f b0; b0.x = q0.x; b0.y = q0.y;
        v2f b1; b1.x = q0.z; b1.y = q0.w;
        v2f b2; b2.x = q1.x; b2.y = q1.y;
        v2f b3; b3.x = q1.z; b3.y = q1.w;
        acc0 = __builtin_amdgcn_wmma_f32_16x16x4_f32(false, a, false, b0, (short)0, acc0, false, false);
        acc1 = __builtin_amdgcn_wmma_f32_16x16x4_f32(false, a, false, b1, (short)0, acc1, false, false);
        acc2 = __builtin_amdgcn_wmma_f32_16x16x4_f32(false, a, false, b2, (short)0, acc2, false, false);
        acc3 = __builtin_amdgcn_wmma_f32_16x16x4_f32(false, a, false, b3, (short)0, acc3, false, false);
    }

    // D layout: VGPR i -> row (i + 8*ksel), col = n0 + j*16 + m
    const int rbase = blockIdx.x * 16 + 8 * ksel;
    const float bi0 = bias[n0 + m];
    const float bi1 = bias[n0 + 16 + m];
    const float bi2 = bias[n0 + 32 + m];
    const float bi3 = bias[n0 + 48 + m];
#pragma unroll
    for (int i = 0; i < 8; ++i) {
        float* o = out + (long)(rbase + i) * HID + n0 + m;
        o[0]  = acc0[i] + bi0;
        o[16] = acc1[i] + bi1;
        o[32] = acc2[i] + bi2;
        o[48] = acc3[i] + bi3;
    }
}

// ------------------------------------------------------------- BatchNorm
__global__ void k_colstats(const float* __restrict__ X, int M, int rowsPerBlock,
                           float* __restrict__ stats /* [0..255]=sum [256..511]=sumsq */) {
    int c  = threadIdx.x;                 // 256 threads = 256 channels
    int r0 = blockIdx.x * rowsPerBlock;
    int r1 = r0 + rowsPerBlock;
    if (r1 > M) r1 = M;
    float s = 0.0f, sq = 0.0f;
    for (int r = r0; r < r1; ++r) {
        float v = X[(long)r * HID + c];
        s += v;
        sq += v * v;
    }
    atomicAdd(&stats[c], s);
    atomicAdd(&stats[HID + c], sq);
}

__global__ void k_bn_finalize(const float* __restrict__ stats, float invM,
                              const float* __restrict__ g, const float* __restrict__ be,
                              float* __restrict__ scale, float* __restrict__ shift) {
    int c = threadIdx.x;
    float mu  = stats[c] * invM;
    float var = stats[HID + c] * invM - mu * mu;   // biased variance
    float sc  = g[c] * rsqrtf(var + BN_EPS);
    scale[c] = sc;
    shift[c] = be[c] - mu * sc;
}

__global__ void k_bnrelu(const float* __restrict__ X, float* __restrict__ Y,
                         const float* __restrict__ scale, const float* __restrict__ shift,
                         int n) {
    int i = blockIdx.x * blockDim.x + threadIdx.x;
    if (i >= n) return;
    int c = i & (HID - 1);
    float v = X[i] * scale[c] + shift[c];
    Y[i] = v > 0.0f ? v : 0.0f;
}

__global__ void k_bn_pool(const float* __restrict__ X, float* __restrict__ H,
                          const float* __restrict__ scale, const float* __restrict__ shift,
                          const int* __restrict__ batch, float* __restrict__ pooled,
                          int layerOff, int n) {
    int i = blockIdx.x * blockDim.x + threadIdx.x;
    if (i >= n) return;
    int c = i & (HID - 1);
    int r = i >> 8;
    float v = X[i] * scale[c] + shift[c];
    H[i] = v;
    atomicAdd(&pooled[(long)batch[r] * (HID * NUM_LAYERS) + layerOff + c], v);
}

// out[128 x 10] = P[128 x 256] @ W2[256 x 10] + b2
__global__ void k_final(const float* __restrict__ P, const float* __restrict__ W2,
                        const float* __restrict__ b2, float* __restrict__ out) {
    int t = blockIdx.x * blockDim.x + threadIdx.x;
    if (t >= NUM_GRAPHS * OUT_CH) return;
    int g = t / OUT_CH;
    int o = t - g * OUT_CH;
    float s = b2[o];
    const float* p = P + g * HID;
#pragma unroll 4
    for (int c = 0; c < HID; ++c) s += p[c] * W2[c * OUT_CH + o];
    out[t] = s;
}

// ---------------------------------------------------------------- launcher
extern "C" void kernel_launch(void* const* d_in, const int* in_sizes, int n_in,
                              void* d_out, int out_size, void* d_ws, size_t ws_size,
                              hipStream_t stream) {
    const float* x     = (const float*)d_in[0];
    const int*   ei    = (const int*)d_in[1];
    const int*   batch = (const int*)d_in[2];
    const int*   esrc  = ei;
    const int*   edst  = ei + N_EDGES;

    // pytree leaf order per layer: W1,W2,b1,b2,be1,be_out,g1,g_out
    auto L = [&](int l, int j) { return (const float*)d_in[3 + l * 8 + j]; };
    const float* Wp1  = (const float*)d_in[35];
    const float* Wp2  = (const float*)d_in[36];
    const float* bp1  = (const float*)d_in[37];
    const float* bp2  = (const float*)d_in[38];
    const float* be_p = (const float*)d_in[39];
    const float* g_p  = (const float*)d_in[40];

    // workspace carve (fp32 elements)
    float*  hbuf   = (float*)d_ws;                      // 50000 x 256
    float*  zbuf   = hbuf + (long)N_NODES * HID;        // 50000 x 256
    float*  t1     = zbuf + (long)N_NODES * HID;        // 50000 x 256
    float*  pooled = t1   + (long)N_NODES * HID;        // 128 x 1024
    float*  pvec   = pooled + NUM_GRAPHS * HID * NUM_LAYERS;  // 128 x 256
    float*  stats  = pvec + NUM_GRAPHS * HID;           // 512
    float*  scale  = stats + 2 * HID;                   // 256
    float*  shift  = scale + HID;                       // 256
    float2* Wq     = (float2*)(shift + HID);            // up to (1024/2)*256 float2 = 1 MB

    const int poolN = NUM_GRAPHS * HID * NUM_LAYERS;    // 131072
    k_zero<<<(poolN + 255) / 256, 256, 0, stream>>>(pooled, poolN);

    for (int l = 0; l < NUM_LAYERS; ++l) {
        const float* h = (l == 0) ? x : hbuf;
        const int C  = (l == 0) ? IN_CH : HID;
        const int nC = N_NODES * C;

        // z = h  then z[dst] += h[src]
        k_copy<<<(nC / 4 + 255) / 256, 256, 0, stream>>>(h, zbuf, nC / 4);
        {
            long tot = (long)N_EDGES * (C / 4);
            k_edge_scatter<<<(unsigned)((tot + 255) / 256), 256, 0, stream>>>(
                h, esrc, edst, zbuf, C / 4);
        }

        // t1 = z @ W1 + b1
        k_packW<<<((C / 2) * HID + 255) / 256, 256, 0, stream>>>(L(l, 0), Wq, (C / 2) * HID);
        if (l == 0)
            k_gemm_wmma<IN_CH><<<N_NODES / 16, 128, 0, stream>>>(zbuf, Wq, L(l, 2), t1);
        else
            k_gemm_wmma<HID><<<N_NODES / 16, 128, 0, stream>>>(zbuf, Wq, L(l, 2), t1);

        // BN1 + ReLU -> zbuf
        k_zero<<<2, 256, 0, stream>>>(stats, 2 * HID);
        k_colstats<<<250, 256, 0, stream>>>(t1, N_NODES, 200, stats);
        k_bn_finalize<<<1, 256, 0, stream>>>(stats, 1.0f / N_NODES, L(l, 6), L(l, 4), scale, shift);
        k_bnrelu<<<(N_NODES * HID + 255) / 256, 256, 0, stream>>>(t1, zbuf, scale, shift, N_NODES * HID);

        // t1 = zbuf @ W2 + b2
        k_packW<<<((HID / 2) * HID + 255) / 256, 256, 0, stream>>>(L(l, 1), Wq, (HID / 2) * HID);
        k_gemm_wmma<HID><<<N_NODES / 16, 128, 0, stream>>>(zbuf, Wq, L(l, 3), t1);

        // outer BN -> hbuf, plus graph pooling
        k_zero<<<2, 256, 0, stream>>>(stats, 2 * HID);
        k_colstats<<<250, 256, 0, stream>>>(t1, N_NODES, 200, stats);
        k_bn_finalize<<<1, 256, 0, stream>>>(stats, 1.0f / N_NODES, L(l, 7), L(l, 5), scale, shift);
        k_bn_pool<<<(N_NODES * HID + 255) / 256, 256, 0, stream>>>(
            t1, hbuf, scale, shift, batch, pooled, l * HID, N_NODES * HID);
    }

    // pool MLP: pvec = pooled @ Wp1 + bp1  (M=128, K=1024)
    k_packW<<<((HID * NUM_LAYERS / 2) * HID + 255) / 256, 256, 0, stream>>>(
        Wp1, Wq, (HID * NUM_LAYERS / 2) * HID);
    k_gemm_wmma<HID * NUM_LAYERS><<<NUM_GRAPHS / 16, 128, 0, stream>>>(pooled, Wq, bp1, pvec);

    k_zero<<<2, 256, 0, stream>>>(stats, 2 * HID);
    k_colstats<<<1, 256, 0, stream>>>(pvec, NUM_GRAPHS, NUM_GRAPHS, stats);
    k_bn_finalize<<<1, 256, 0, stream>>>(stats, 1.0f / NUM_GRAPHS, g_p, be_p, scale, shift);
    k_bnrelu<<<(NUM_GRAPHS * HID + 255) / 256, 256, 0, stream>>>(pvec, pvec, scale, shift,
                                                                 NUM_GRAPHS * HID);

    k_final<<<(NUM_GRAPHS * OUT_CH + 127) / 128, 128, 0, stream>>>(pvec, Wp2, bp2, (float*)d_out);
}